// TriangleOscillator_59425167508144
// MI455X (gfx1250) — compile-verified
//
#include <hip/hip_runtime.h>
#include <hip/hip_bf16.h>
#include <math.h>

// ---------------------------------------------------------------------------
// Triangle oscillator bank, MI455X (gfx1250, wave32).
//   out[b,t] = sum_p amp_p * sin(2*pi * cumsum_t((f[b,t]*m_p/SR) mod 1))
// Factorization: phase_p(t) (in revolutions) == m_p * R(t) + integer,
// R = cumsum(f/SR); integer parts drop under sin. 3-pass hierarchical scan:
//   pass1: per-chunk partial sums of f/SR
//   pass2: f64 exclusive scan of chunk sums; wrapped per-harmonic carries
//   pass3: TDM chunk->LDS, wave32 shuffle scan, 32-harmonic unrolled FMA+fract+v_sin
// Workload is TRANS/VALU-bound (~67M v_sin_f32); HBM traffic is only ~24MB.
// No matmul structure -> WMMA inapplicable; CDNA5 paths used: TDM async tensor
// load + s_wait_tensorcnt, wave32 shuffle scan, v_sin/v_fract TRANS co-exec.
// ---------------------------------------------------------------------------

#define BATCH   16
#define TIME    131072
#define NPITCH  32
#define CHUNK   512
#define NCHUNK  (TIME / CHUNK)      // 256

#define SR      44100.0f
#define INV_SR  (1.0f / 44100.0f)
#define NYQ     22050.0f
#define TRI_C   0.8105694691387023f  // 8/pi^2

typedef unsigned int u32;
typedef unsigned long long u64;
typedef __attribute__((ext_vector_type(4))) u32 v4u;
typedef __attribute__((ext_vector_type(8))) int v8i;
typedef __attribute__((ext_vector_type(4))) int v4i;

// ---------------- pass 3 (defined first so it leads the disasm) -------------
__global__ void tri_main(const float* __restrict__ f,
                         const float* __restrict__ carry,
                         float* __restrict__ out) {
    const int c = blockIdx.x, b = blockIdx.y;
    const int lane = threadIdx.x;                     // 32 threads = 1 wave
    __shared__ float buf[CHUNK];

    const float* gsrc = f + (size_t)b * TIME + (size_t)c * CHUNK;

    // per-harmonic wrapped phase carry; lane p holds carry_p.
    // Issued before the tensor-wait so its latency overlaps the DMA.
    const float cvec = carry[((size_t)b * NCHUNK + c) * NPITCH + lane];

#if __has_builtin(__builtin_amdgcn_tensor_load_to_lds)
    // ---- Tensor Data Mover: DMA the chunk (512 f32 = 2KB) into LDS ----
    const u32 lds = (u32)(uintptr_t)&buf[0];          // LDS byte offset (addr[31:0])
    const u64 ga  = (u64)(uintptr_t)gsrc;
    v4u g0;
    g0[0] = 1u;                                       // count=1, user mode, no gather
    g0[1] = lds;                                      // lds_addr
    g0[2] = (u32)ga;                                  // global_addr[31:0]
    g0[3] = (u32)((ga >> 32) & 0x1FFFFFFull) | 0x80000000u; // ga[56:32] | type=2
    v8i g1;
    g1[0] = (int)(2u << 16);                          // data_size=2 -> 4 bytes
    g1[1] = (int)((u32)CHUNK << 16);                  // tensor_dim0[15:0]=512
    g1[2] = (int)(1u << 16);                          // tensor_dim0 hi=0, tensor_dim1=1
    g1[3] = (int)((u32)CHUNK << 16);                  // tile_dim0=512
    g1[4] = 0;                                        // tile_dim1=0, tile_dim2=0
    g1[5] = (int)CHUNK;                               // tensor_dim0_stride=512
    g1[6] = 0;
    g1[7] = 0;
    v4i gz4 = {0, 0, 0, 0};
    v8i gz8 = {0, 0, 0, 0, 0, 0, 0, 0};
    // amdgpu-toolchain (clang-23 / therock-10.0) 6-arg form
    __builtin_amdgcn_tensor_load_to_lds(g0, g1, gz4, gz4, gz8, 0);
    __builtin_amdgcn_s_wait_tensorcnt(0);
    __asm__ volatile("" ::: "memory");
#else
    // Host pass / toolchains without the TDM builtin: plain coalesced staging.
#pragma unroll
    for (int i = 0; i < CHUNK / 32; ++i)
        buf[i * 32 + lane] = gsrc[i * 32 + lane];
    __syncthreads();
#endif

    float* outp = out + (size_t)b * TIME + (size_t)c * CHUNK;

    float base = 0.0f;                                // chunk-local scan base
    for (int j = 0; j < CHUNK / 32; ++j) {
        const float fv = buf[j * 32 + lane];
        float x = fv * INV_SR;                        // revolutions/sample (m=1)
        // wave32 inclusive scan (5 shuffle-add steps)
#pragma unroll
        for (int d = 1; d < 32; d <<= 1) {
            float y = __shfl_up(x, d, 32);
            x += (lane >= d) ? y : 0.0f;
        }
        const float X = base + x;                     // chunk-local cumsum (<= ~3.5)
        base += __shfl(x, 31, 32);

        float acc = 0.0f;
#pragma unroll
        for (int p = 0; p < NPITCH; ++p) {
            const float MP  = 2.0f * (float)p + 1.0f;
            const float AMP = ((p & 1) ? -TRI_C : TRI_C) / (MP * MP);
            const float cb  = __shfl(cvec, p, 32);    // v_readlane broadcast
            float ph = __builtin_amdgcn_fractf(fmaf(MP, X, cb)); // [0,1) revs
            float s  = __builtin_amdgcn_sinf(ph);     // v_sin_f32: sin(2*pi*ph)
            s = (fv * MP < NYQ) ? s : 0.0f;           // Nyquist mask
            acc = fmaf(AMP, s, acc);
        }
        outp[j * 32 + lane] = acc;                    // coalesced b32 store
    }
}

// ---------------- pass 1: per-chunk sums of f (scaled by 1/SR) --------------
__global__ void tri_chunksum(const float* __restrict__ f, float* __restrict__ S) {
    const int c = blockIdx.x, b = blockIdx.y;
    const int lane = threadIdx.x;                     // 32 threads = 1 wave
    const float* src = f + (size_t)b * TIME + (size_t)c * CHUNK;
    float s = 0.0f;
#pragma unroll
    for (int i = 0; i < CHUNK / 32; ++i)
        s += src[i * 32 + lane];
    s *= INV_SR;
#pragma unroll
    for (int d = 16; d >= 1; d >>= 1)
        s += __shfl_xor(s, d, 32);
    if (lane == 0) S[b * NCHUNK + c] = s;
}

// ------ pass 2: f64 exclusive scan per batch row; wrapped harmonic carries --
__global__ void tri_scan(const float* __restrict__ S, float* __restrict__ carry) {
    const int tid = threadIdx.x;          // 512 threads: (b,p)
    const int b = tid >> 5;
    const int p = tid & 31;
    const double m = (double)(2 * p + 1);
    double C = 0.0;
    for (int c = 0; c < NCHUNK; ++c) {
        double ph = m * C;
        carry[((size_t)b * NCHUNK + c) * NPITCH + p] = (float)(ph - floor(ph));
        C += (double)S[b * NCHUNK + c];
    }
}

// ---------------------------------------------------------------------------
extern "C" void kernel_launch(void* const* d_in, const int* in_sizes, int n_in,
                              void* d_out, int out_size, void* d_ws, size_t ws_size,
                              hipStream_t stream) {
    const float* f = (const float*)d_in[0];
    float* out = (float*)d_out;

    // workspace: chunk sums (16KB) then carries (512KB)
    float* S = (float*)d_ws;
    float* carry = S + BATCH * NCHUNK;

    tri_chunksum<<<dim3(NCHUNK, BATCH), 32, 0, stream>>>(f, S);
    tri_scan<<<1, BATCH * NPITCH, 0, stream>>>(S, carry);
    tri_main<<<dim3(NCHUNK, BATCH), 32, 0, stream>>>(f, carry, out);
}